// SGRUCell_17583596109967
// MI455X (gfx1250) — compile-verified
//
#include <hip/hip_runtime.h>
#include <math.h>

#define TT 64
#define BB 8
#define INN 256
#define HH 512
#define MM 32
#define H3 (3*HH)
#define BH (BB*HH)
#define BHH ((size_t)BB*HH*HH)
#define CLIPV 1.0f

typedef float v2f __attribute__((ext_vector_type(2)));
typedef float v8f __attribute__((ext_vector_type(8)));

// ---- output layout (flat, reference return order) ----
#define OFF_V    ((size_t)0)
#define OFF_H    (OFF_V + BH)
#define OFF_DU   (OFF_H + BH)
#define OFF_TE   (OFF_DU + BHH)
#define OFF_TEE  (OFF_TE + BH)
#define OFF_OUTS (OFF_TEE + BHH)
#define OFF_DUS  (OFF_OUTS + (size_t)TT*BH)
#define OFF_MODS (OFF_DUS + (size_t)TT*BHH)
#define OFF_SS   (OFF_MODS + (size_t)TT*BB*3*MM)
#define OFF_MS   (OFF_SS + (size_t)TT*BB)
#define OFF_RS   (OFF_MS + (size_t)TT*BB)

// ---- workspace layout (floats) ----
#define WS_SPA 0
#define WS_TAU 1
#define WS_S   2            // 8 per-batch s
#define WS_M   10           // 8 per-batch m
#define WS_TE0 32
#define WS_TE1 (32 + BH)
#define WS_HFW (32 + 2*BH)
#define WS_FW  (32 + 3*BH)
#define WS_HB  (32 + 4*BH)
#define WS_VB  (32 + 5*BH)

__device__ __forceinline__ float sigm(float x) { return 1.f / (1.f + expf(-x)); }

// ---------------- init ----------------
__global__ void init_kernel(const float* alpha, const float* tauU,
                            const float* h0, const float* v0,
                            const float* te0, const float* tE0,
                            float* ws, float* out) {
  size_t idx = (size_t)blockIdx.x * blockDim.x + threadIdx.x;
  if (idx == 0) {
    float a = alpha[0];
    ws[WS_SPA] = (a > 0.f) ? (a + log1pf(expf(-a))) : log1pf(expf(a)); // softplus
    ws[WS_TAU] = 1.f / (1.f + expf(-tauU[0]));                         // sigmoid
  }
  size_t stride = (size_t)gridDim.x * blockDim.x;
  for (size_t i = idx; i < BHH; i += stride) {
    out[OFF_TEE + i] = tE0[i];     // live tE state resides in output slot
    if (i < BH) {
      ws[WS_TE0 + i] = te0[i];
      ws[WS_FW + i]  = 0.f;        // fw_0 = sp_a * dU0 @ h0 = 0
      ws[WS_HB + i]  = h0[i];
      ws[WS_VB + i]  = v0[i];
    }
  }
}

// ---------------- gate kernel (1 block, 1024 threads = 32 waves) ----------------
struct GateArgs {
  const float *x, *x2h_w, *x2h_b, *h2h_w, *h2h_b;
  const float *lnx_g, *lnx_b, *lnh_g, *lnh_b;
  const float *h2mod_w, *h2mod_b;
  const float *mod2hr_w, *mod2hr_b, *mod2hs_w, *mod2hs_b, *mod2hm_w, *mod2hm_b;
  float *ws, *out;
  int t;
};

__global__ __launch_bounds__(1024) void gate_kernel(GateArgs A) {
  __shared__ float sm[36864];        // 144 KB (< 320 KB WGP LDS)
  const int tid  = threadIdx.x;
  const int lane = tid & 31;
  const int wv   = tid >> 5;
  float* ws  = A.ws;
  float* out = A.out;
  const int t = A.t;

  float* sX   = sm;                 // [16][256], rows 8-15 zero-padded
  float* sH   = sm + 4096;          // [16][512], rows 8-15 zero-padded
  float* sWx  = sm + 12288;         // [8][1536]
  float* sWh  = sm + 24576;         // [8][1536]
  float* sNH  = sm;                 // alias: [8][512] new_h (sX dead)
  float* sMod = sm + 4096;          // alias: [8][96]  (sH dead)
  float* sScl = sm + 4096 + 800;    // [8] r

  const float* xt = A.x + (size_t)t * BB * INN;
  float* hbuf = ws + WS_HB;
  for (int i = tid; i < 16 * INN; i += 1024) sX[i] = (i < BB * INN) ? xt[i] : 0.f;
  for (int i = tid; i < 16 * HH;  i += 1024) sH[i] = (i < BH) ? hbuf[i] : 0.f;
  __syncthreads();

  // ---- WMMA f32 16x16x4 GEMMs: C[16,16] += A[16,4] * B[4,16], batch padded to 16 rows ----
  // A layout: lanes 0-15 M=0..15 (K=k0,k0+1), lanes 16-31 M=0..15 (K=k0+2,k0+3)
  // B layout: lanes 0-15 N (K=k0,k0+1), lanes 16-31 N (K=k0+2,k0+3); B[k][n] = W[n][k]
  {
    const int m  = lane & 15;
    const int kh = lane >> 4;
    for (int g = 0; g < 2; ++g) {
      const float* Wmat = g ? A.h2h_w : A.x2h_w;
      const float* sA   = g ? sH : sX;
      float* sC         = g ? sWh : sWx;
      const int K       = g ? HH : INN;
      for (int tt2 = 0; tt2 < 3; ++tt2) {
        int nb = (wv + tt2 * 32) * 16;
        v8f c = {};
        const float* wbase = Wmat + (size_t)(nb + m) * K;
        const float* abase = sA + m * K + 2 * kh;
        for (int k0 = 0; k0 < K; k0 += 4) {
          v2f a; a[0] = abase[k0];      a[1] = abase[k0 + 1];
          v2f b; b[0] = wbase[k0 + 2 * kh]; b[1] = wbase[k0 + 2 * kh + 1];
          c = __builtin_amdgcn_wmma_f32_16x16x4_f32(false, a, false, b,
                                                    (short)0, c, false, false);
        }
        if (lane < 16) {
          #pragma unroll
          for (int r = 0; r < 8; ++r) sC[r * H3 + nb + lane] = c[r]; // rows 0..7 valid
        }
      }
    }
  }
  __syncthreads();

  // ---- LayerNorm: waves 0-7 -> Wx rows, waves 8-15 -> Wh rows (+bias, +fw on dv part) ----
  if (wv < BB) {
    float* row = sWx + wv * H3;
    float sum = 0.f, sq = 0.f;
    for (int n = lane; n < H3; n += 32) {
      float v = row[n] + A.x2h_b[n];
      row[n] = v; sum += v; sq += v * v;
    }
    for (int o = 16; o; o >>= 1) { sum += __shfl_xor(sum, o, 32); sq += __shfl_xor(sq, o, 32); }
    float mu  = sum * (1.f / H3);
    float inv = rsqrtf(sq * (1.f / H3) - mu * mu + 1e-5f);
    for (int n = lane; n < H3; n += 32)
      row[n] = (row[n] - mu) * inv * A.lnx_g[n] + A.lnx_b[n];
  } else if (wv < 2 * BB) {
    const int m2 = wv - BB;
    float* row = sWh + m2 * H3;
    const float* fw = ws + WS_FW + m2 * HH;
    float sum = 0.f, sq = 0.f;
    for (int n = lane; n < H3; n += 32) {
      float v = row[n] + A.h2h_b[n];
      if (n >= 2 * HH) v += fw[n - 2 * HH];
      row[n] = v; sum += v; sq += v * v;
    }
    for (int o = 16; o; o >>= 1) { sum += __shfl_xor(sum, o, 32); sq += __shfl_xor(sq, o, 32); }
    float mu  = sum * (1.f / H3);
    float inv = rsqrtf(sq * (1.f / H3) - mu * mu + 1e-5f);
    for (int n = lane; n < H3; n += 32)
      row[n] = (row[n] - mu) * inv * A.lnh_g[n] + A.lnh_b[n];
  }
  __syncthreads();

  // ---- gates, v/h update ----
  {
    float* vbuf = ws + WS_VB;
    float* hfw  = ws + WS_HFW;
    #pragma unroll
    for (int kk = 0; kk < 4; ++kk) {
      int idx = tid + kk * 1024;        // idx = m*512 + i
      int m = idx >> 9, i = idx & 511;
      float z  = sigm(sWx[m * H3 + i]          + sWh[m * H3 + i]);
      float o  = sigm(sWx[m * H3 + HH + i]     + sWh[m * H3 + HH + i]);
      float dv =       sWx[m * H3 + 2 * HH + i] + sWh[m * H3 + 2 * HH + i];
      float vn = (1.f - z) * vbuf[idx] + z * dv;
      vbuf[idx] = vn;
      float nh = fmaxf(vn, 0.f);
      sNH[idx]  = nh;
      hbuf[idx] = nh;
      out[OFF_OUTS + (size_t)t * BH + idx] = nh;
      hfw[idx] = o * nh;
    }
  }
  __syncthreads();

  // ---- mod = relu(new_h @ h2mod_w^T + b): 768 small dot products ----
  if (tid < BB * 3 * MM) {
    int b = tid / (3 * MM), j = tid % (3 * MM);
    const float* wrow = A.h2mod_w + (size_t)j * HH;
    const float* hn = sNH + b * HH;
    float acc = A.h2mod_b[j];
    for (int k2 = 0; k2 < HH; ++k2) acc = fmaf(hn[k2], wrow[k2], acc);
    float mv = fmaxf(acc, 0.f);
    sMod[tid] = mv;
    out[OFF_MODS + (size_t)t * BB * 3 * MM + tid] = mv;
  }
  __syncthreads();

  // ---- per-batch scalars r, s, m ----
  if (tid < BB) {
    int b = tid;
    float r = A.mod2hr_b[0], s = A.mod2hs_b[0], mm2 = A.mod2hm_b[0];
    for (int k2 = 0; k2 < MM; ++k2) {
      r   += sMod[b * 96 + k2]          * A.mod2hr_w[k2];
      s   += sMod[b * 96 + MM + k2]     * A.mod2hs_w[k2];
      mm2 += sMod[b * 96 + 2 * MM + k2] * A.mod2hm_w[k2];
    }
    r = sigm(r); s = sigm(s);
    float mval = mm2 - tanhf(mm2);
    out[OFF_RS + (size_t)t * BB + b] = r;
    out[OFF_SS + (size_t)t * BB + b] = s;
    out[OFF_MS + (size_t)t * BB + b] = mval;
    ws[WS_S + b] = s; ws[WS_M + b] = mval;
    sScl[b] = r;
  }
  __syncthreads();

  // ---- te update (double-buffered; plastic kernel still needs te_old) ----
  {
    float* teOld = ws + ((t & 1) ? WS_TE1 : WS_TE0);
    float* teNew = ws + ((t & 1) ? WS_TE0 : WS_TE1);
    const float* hfw = ws + WS_HFW;
    #pragma unroll
    for (int kk = 0; kk < 4; ++kk) {
      int idx = tid + kk * 1024;
      float r = sScl[idx >> 9];
      teNew[idx] = (1.f - r) * teOld[idx] + r * hfw[idx];
    }
  }
}

// ---------------- plasticity kernel: one wave per (b,i) row ----------------
__global__ __launch_bounds__(128) void plastic_kernel(const float* h2h_w, const float* dU0,
                                                      float* ws, float* out, int t) {
  const int lane = threadIdx.x & 31;
  const int w = blockIdx.x * 4 + (threadIdx.x >> 5);   // 0..4095
  const int b = w >> 9, i = w & 511;
  const float sp_a = ws[WS_SPA], tau = ws[WS_TAU];
  const float s = ws[WS_S + b], mval = ws[WS_M + b];
  const float* teOld = ws + ((t & 1) ? WS_TE1 : WS_TE0);
  const float* hfw   = ws + WS_HFW;
  const float* hnew  = ws + WS_HB;
  const float hfw_i = hfw[b * HH + i];
  const float te_i  = teOld[b * HH + i];
  const float* dUold = (t == 0)
      ? (dU0 + (size_t)b * HH * HH + (size_t)i * HH)
      : (out + OFF_DUS + (size_t)(t - 1) * BHH + (size_t)b * HH * HH + (size_t)i * HH);
  float* dUnew = out + OFF_DUS + (size_t)t * BHH + (size_t)b * HH * HH + (size_t)i * HH;
  float* tE    = out + OFF_TEE + (size_t)b * HH * HH + (size_t)i * HH;
  const float* wrow = h2h_w + (size_t)(2 * HH + i) * HH;   // W_dv row i
  const float inv_a = 1.f / (sp_a + 1e-8f);
  float acc = 0.f;
  for (int j = lane; j < HH; j += 32) {
    float outer = hfw_i * teOld[b * HH + j] - te_i * hfw[b * HH + j];
    float tEn = (1.f - s) * tE[j] + s * outer;
    tE[j] = tEn;
    float d = (1.f - tau) * dUold[j] + tau * mval * tEn;
    float wv2 = wrow[j];
    float up = fmaxf(CLIPV - wv2, 0.f) * inv_a;
    float lo = -fmaxf(CLIPV + wv2, 0.f) * inv_a;
    d = fminf(d, up);
    d = fmaxf(d, lo);
    dUnew[j] = d;
    acc = fmaf(d, hnew[b * HH + j], acc);     // fw for next step, fused
  }
  for (int o = 16; o; o >>= 1) acc += __shfl_xor(acc, o, 32);
  if (lane == 0) ws[WS_FW + b * HH + i] = sp_a * acc;
}

// ---------------- finalize ----------------
__global__ void final_kernel(const float* ws, float* out) {
  size_t idx = (size_t)blockIdx.x * blockDim.x + threadIdx.x;
  size_t stride = (size_t)gridDim.x * blockDim.x;
  for (size_t i = idx; i < BHH; i += stride) {
    out[OFF_DU + i] = out[OFF_DUS + (size_t)(TT - 1) * BHH + i];
    if (i < BH) {
      out[OFF_V + i]  = ws[WS_VB + i];
      out[OFF_H + i]  = ws[WS_HB + i];
      out[OFF_TE + i] = ws[WS_TE0 + i];   // T even -> final te lands in buffer 0
    }
  }
}

extern "C" void kernel_launch(void* const* d_in, const int* in_sizes, int n_in,
                              void* d_out, int out_size, void* d_ws, size_t ws_size,
                              hipStream_t stream) {
  (void)in_sizes; (void)n_in; (void)out_size; (void)ws_size;
  const float* x        = (const float*)d_in[0];
  const float* h0       = (const float*)d_in[1];
  const float* v0       = (const float*)d_in[2];
  const float* dU0      = (const float*)d_in[3];
  const float* te0      = (const float*)d_in[4];
  const float* tE0      = (const float*)d_in[5];
  const float* x2h_w    = (const float*)d_in[6];
  const float* x2h_b    = (const float*)d_in[7];
  const float* h2h_w    = (const float*)d_in[8];
  const float* h2h_b    = (const float*)d_in[9];
  const float* lnx_g    = (const float*)d_in[10];
  const float* lnx_b    = (const float*)d_in[11];
  const float* lnh_g    = (const float*)d_in[12];
  const float* lnh_b    = (const float*)d_in[13];
  const float* h2mod_w  = (const float*)d_in[14];
  const float* h2mod_b  = (const float*)d_in[15];
  const float* mod2hr_w = (const float*)d_in[16];
  const float* mod2hr_b = (const float*)d_in[17];
  const float* mod2hs_w = (const float*)d_in[18];
  const float* mod2hs_b = (const float*)d_in[19];
  const float* mod2hm_w = (const float*)d_in[20];
  const float* mod2hm_b = (const float*)d_in[21];
  const float* alpha    = (const float*)d_in[22];
  const float* tauU     = (const float*)d_in[23];

  float* ws  = (float*)d_ws;
  float* out = (float*)d_out;

  init_kernel<<<4096, 512, 0, stream>>>(alpha, tauU, h0, v0, te0, tE0, ws, out);

  GateArgs ga;
  ga.x = x; ga.x2h_w = x2h_w; ga.x2h_b = x2h_b; ga.h2h_w = h2h_w; ga.h2h_b = h2h_b;
  ga.lnx_g = lnx_g; ga.lnx_b = lnx_b; ga.lnh_g = lnh_g; ga.lnh_b = lnh_b;
  ga.h2mod_w = h2mod_w; ga.h2mod_b = h2mod_b;
  ga.mod2hr_w = mod2hr_w; ga.mod2hr_b = mod2hr_b;
  ga.mod2hs_w = mod2hs_w; ga.mod2hs_b = mod2hs_b;
  ga.mod2hm_w = mod2hm_w; ga.mod2hm_b = mod2hm_b;
  ga.ws = ws; ga.out = out;

  for (int t = 0; t < TT; ++t) {
    ga.t = t;
    gate_kernel<<<1, 1024, 0, stream>>>(ga);
    plastic_kernel<<<1024, 128, 0, stream>>>(h2h_w, dU0, ws, out, t);
  }
  final_kernel<<<4096, 512, 0, stream>>>(ws, out);
}